// GptOssExpertsLinear_137438954363
// MI455X (gfx1250) — compile-verified
//
#include <hip/hip_runtime.h>

// ---------------------------------------------------------------------------
// GPT-OSS MoE experts (E=16, H=D=2880, T=2048), fp32, MI455X / gfx1250.
// HBM-bound (1.59 GB weights vs 1.63 TFLOP) -> keep exact fp32 precision with
// V_WMMA_F32_16X16X4_F32. Per-expert kernels keep that expert's ~100 MB of
// weights L2-resident. Weight tiles are staged into LDS with
// GLOBAL_LOAD_ASYNC_TO_LDS_B128 (ASYNCcnt) to kill the 4x per-wave redundant
// B loads; fragments come from bank-conflict-free ds_load_b32.
// ---------------------------------------------------------------------------

typedef __attribute__((ext_vector_type(2))) float v2f;
typedef __attribute__((ext_vector_type(8))) float v8f;

#define NUM_E 16
#define DIM_H 2880
#define DIM_D 2880
#define NUM_T 2048
#define GELU_ALPHA 1.702f
#define CLIP_LIMIT 7.0f

#define KC 32          // K rows staged per LDS chunk
#define LDS_STRIDE 80  // padded row stride (floats); 2*LDS_STRIDE mod 64 == 32
                       // -> the two half-wave K-rows of a fragment hit
                       //    disjoint bank groups (conflict-free ds_load_b32)

// Per-lane async DMA: copy 16 bytes global -> LDS (tracked by ASYNCcnt).
// Generic pointers to __shared__ carry the LDS offset in their low 32 bits.
__device__ __forceinline__ void async_copy16(unsigned int lds_addr, const float* gp) {
  unsigned long long ga = (unsigned long long)(uintptr_t)gp;
  asm volatile("global_load_async_to_lds_b128 %0, %1, off"
               :: "v"(lds_addr), "v"(ga) : "memory");
}
__device__ __forceinline__ void wait_async() {
  asm volatile("s_wait_asynccnt 0x0" ::: "memory");
}

__global__ __launch_bounds__(256) void zero_kernel(float* __restrict__ p, long long n) {
  long long i = (long long)blockIdx.x * 256 + threadIdx.x;
  if (i < n) p[i] = 0.0f;
}

// ---------------------------------------------------------------------------
// Kernel 1 (per expert): inter[t,d] = (clip(up)+1) * gate*sigmoid(ALPHA*gate)
// grid = (D/64, T/128), block = 128 (4 waves, M-stacked).
// Wave tile: 32M x 64N -> 2x4 sub-tiles of 16x16, K-chunks of 32 via LDS.
// ---------------------------------------------------------------------------
__global__ __launch_bounds__(128)
void gateup_kernel(const float* __restrict__ x,     // [T, H]
                   const float* __restrict__ Wg,    // [H, D] (expert slice)
                   const float* __restrict__ bg,    // [D]
                   const float* __restrict__ Wu,    // [H, D]
                   const float* __restrict__ bu,    // [D]
                   float* __restrict__ inter) {     // [T, D]
  __shared__ float sG[KC * LDS_STRIDE];
  __shared__ float sU[KC * LDS_STRIDE];

  const int t    = (int)threadIdx.x;
  const int lane = t & 31;
  const int wave = t >> 5;
  const int lo   = lane & 15;   // M (A) / N (B,C) position
  const int hi   = lane >> 4;   // K-half select for A/B, +8 rows for C
  const int baseN = blockIdx.x * 64;
  const int baseM = blockIdx.y * 128 + wave * 32;

  v8f cg[2][4], cu[2][4];
  const v8f vzero = {0.f, 0.f, 0.f, 0.f, 0.f, 0.f, 0.f, 0.f};
#pragma unroll
  for (int i = 0; i < 2; ++i)
#pragma unroll
    for (int j = 0; j < 4; ++j) { cg[i][j] = vzero; cu[i][j] = vzero; }

  const float* xp0 = x + (long long)(baseM + lo) * DIM_H;
  const float* xp1 = x + (long long)(baseM + 16 + lo) * DIM_H;

  for (int k0 = 0; k0 < DIM_H; k0 += KC) {
    __syncthreads();  // previous chunk's fragment reads are done
    // Stage B tiles [KC x 64] for Wg and Wu: 128 threads x 4 rounds x 16 B.
#pragma unroll
    for (int rr = 0; rr < 4; ++rr) {
      const int idx = rr * 128 + t;          // 0..511 float4 slots
      const int row = idx >> 4;              // 0..31 (K row in chunk)
      const int col = (idx & 15) << 2;       // 0..60 (N col, float4-aligned)
      const long long go = (long long)(k0 + row) * DIM_D + baseN + col;
      const unsigned lg = (unsigned)(uintptr_t)(sG + row * LDS_STRIDE + col);
      const unsigned lu = (unsigned)(uintptr_t)(sU + row * LDS_STRIDE + col);
      async_copy16(lg, Wg + go);
      async_copy16(lu, Wu + go);
      if (k0 + KC < DIM_H) {  // pull next chunk HBM -> L2 early
        __builtin_prefetch((const void*)(Wg + go + (long long)KC * DIM_D), 0, 1);
        __builtin_prefetch((const void*)(Wu + go + (long long)KC * DIM_D), 0, 1);
      }
    }
    wait_async();
    __syncthreads();  // tiles visible to all waves

#pragma unroll
    for (int ks = 0; ks < KC; ks += 4) {
      const int kb = ks + 2 * hi;                    // this lane's K pair
      const v2f a0 = *(const v2f*)(xp0 + k0 + kb);   // A frag, M sub-tile 0
      const v2f a1 = *(const v2f*)(xp1 + k0 + kb);   // A frag, M sub-tile 1
#pragma unroll
      for (int j = 0; j < 4; ++j) {
        const int s = kb * LDS_STRIDE + j * 16 + lo;
        v2f bgf, buf;
        bgf.x = sG[s]; bgf.y = sG[s + LDS_STRIDE];
        buf.x = sU[s]; buf.y = sU[s + LDS_STRIDE];
        cg[0][j] = __builtin_amdgcn_wmma_f32_16x16x4_f32(false, a0, false, bgf,
                                                         (short)0, cg[0][j], false, false);
        cg[1][j] = __builtin_amdgcn_wmma_f32_16x16x4_f32(false, a1, false, bgf,
                                                         (short)0, cg[1][j], false, false);
        cu[0][j] = __builtin_amdgcn_wmma_f32_16x16x4_f32(false, a0, false, buf,
                                                         (short)0, cu[0][j], false, false);
        cu[1][j] = __builtin_amdgcn_wmma_f32_16x16x4_f32(false, a1, false, buf,
                                                         (short)0, cu[1][j], false, false);
      }
    }
  }

  // Epilogue: bias, clip, glu, write inter. C layout: n = lo, m = r + 8*hi.
#pragma unroll
  for (int i = 0; i < 2; ++i) {
#pragma unroll
    for (int j = 0; j < 4; ++j) {
      const int n = baseN + j * 16 + lo;
      const float bgv = bg[n];
      const float buv = bu[n];
#pragma unroll
      for (int r = 0; r < 8; ++r) {
        const int m = baseM + i * 16 + r + 8 * hi;
        float g = cg[i][j][r] + bgv;
        g = fminf(g, CLIP_LIMIT);
        float u = cu[i][j][r] + buv;
        u = fminf(fmaxf(u, -CLIP_LIMIT), CLIP_LIMIT);
        const float glu = g / (1.0f + __expf(-GELU_ALPHA * g));  // g*sigmoid(a*g)
        inter[(long long)m * DIM_D + n] = (u + 1.0f) * glu;
      }
    }
  }
}

// ---------------------------------------------------------------------------
// Kernel 2 (per expert): out[t,h] += rw[t,e] * (inter @ Wd + bd)
// grid = (H/64, T/128), block = 128. Same tiling, one staged matrix.
// ---------------------------------------------------------------------------
__global__ __launch_bounds__(128)
void down_kernel(const float* __restrict__ inter,  // [T, D]
                 const float* __restrict__ Wd,     // [D, H] (expert slice)
                 const float* __restrict__ bd,     // [H]
                 const float* __restrict__ rw,     // [T, E]
                 int e,
                 float* __restrict__ out) {        // [T, H]
  __shared__ float sW[KC * LDS_STRIDE];

  const int t    = (int)threadIdx.x;
  const int lane = t & 31;
  const int wave = t >> 5;
  const int lo   = lane & 15;
  const int hi   = lane >> 4;
  const int baseN = blockIdx.x * 64;
  const int baseM = blockIdx.y * 128 + wave * 32;

  v8f c[2][4];
  const v8f vzero = {0.f, 0.f, 0.f, 0.f, 0.f, 0.f, 0.f, 0.f};
#pragma unroll
  for (int i = 0; i < 2; ++i)
#pragma unroll
    for (int j = 0; j < 4; ++j) c[i][j] = vzero;

  const float* ap0 = inter + (long long)(baseM + lo) * DIM_D;
  const float* ap1 = inter + (long long)(baseM + 16 + lo) * DIM_D;

  for (int k0 = 0; k0 < DIM_D; k0 += KC) {
    __syncthreads();
#pragma unroll
    for (int rr = 0; rr < 4; ++rr) {
      const int idx = rr * 128 + t;
      const int row = idx >> 4;
      const int col = (idx & 15) << 2;
      const long long go = (long long)(k0 + row) * DIM_H + baseN + col;
      const unsigned lw = (unsigned)(uintptr_t)(sW + row * LDS_STRIDE + col);
      async_copy16(lw, Wd + go);
      if (k0 + KC < DIM_D) {
        __builtin_prefetch((const void*)(Wd + go + (long long)KC * DIM_H), 0, 1);
      }
    }
    wait_async();
    __syncthreads();

#pragma unroll
    for (int ks = 0; ks < KC; ks += 4) {
      const int kb = ks + 2 * hi;
      const v2f a0 = *(const v2f*)(ap0 + k0 + kb);
      const v2f a1 = *(const v2f*)(ap1 + k0 + kb);
#pragma unroll
      for (int j = 0; j < 4; ++j) {
        const int s = kb * LDS_STRIDE + j * 16 + lo;
        v2f bf;
        bf.x = sW[s]; bf.y = sW[s + LDS_STRIDE];
        c[0][j] = __builtin_amdgcn_wmma_f32_16x16x4_f32(false, a0, false, bf,
                                                        (short)0, c[0][j], false, false);
        c[1][j] = __builtin_amdgcn_wmma_f32_16x16x4_f32(false, a1, false, bf,
                                                        (short)0, c[1][j], false, false);
      }
    }
  }

#pragma unroll
  for (int i = 0; i < 2; ++i) {
#pragma unroll
    for (int j = 0; j < 4; ++j) {
      const int n = baseN + j * 16 + lo;
      const float bdv = bd[n];
#pragma unroll
      for (int r = 0; r < 8; ++r) {
        const int m = baseM + i * 16 + r + 8 * hi;
        const float w = rw[(long long)m * NUM_E + e];
        out[(long long)m * DIM_H + n] += w * (c[i][j][r] + bdv);
      }
    }
  }
}

// ---------------------------------------------------------------------------
// Launcher. Inputs (setup_inputs order):
//   0: hidden_states [2,1024,2880] f32   1: routing_weights [2048,16] f32
//   2: Wg [16,2880,2880]  3: bg [16,2880]  4: Wu  5: bu
//   6: Wd [16,2880,2880]  7: bd [16,2880]
// d_out: [2048,2880] f32.  d_ws: T*D*4 = 23.6 MB for inter.
// ---------------------------------------------------------------------------
extern "C" void kernel_launch(void* const* d_in, const int* in_sizes, int n_in,
                              void* d_out, int out_size, void* d_ws, size_t ws_size,
                              hipStream_t stream) {
  const float* x  = (const float*)d_in[0];
  const float* rw = (const float*)d_in[1];
  const float* Wg = (const float*)d_in[2];
  const float* bg = (const float*)d_in[3];
  const float* Wu = (const float*)d_in[4];
  const float* bu = (const float*)d_in[5];
  const float* Wd = (const float*)d_in[6];
  const float* bd = (const float*)d_in[7];
  float* out = (float*)d_out;
  float* ws  = (float*)d_ws;

  const long long nOut = (long long)NUM_T * DIM_H;
  zero_kernel<<<(int)((nOut + 255) / 256), 256, 0, stream>>>(out, nOut);

  const dim3 grid1(DIM_D / 64, NUM_T / 128);  // (45, 16)
  const dim3 grid2(DIM_H / 64, NUM_T / 128);  // (45, 16)

  for (int e = 0; e < NUM_E; ++e) {
    const long long wOff = (long long)e * DIM_H * DIM_D;
    gateup_kernel<<<grid1, 128, 0, stream>>>(
        x, Wg + wOff, bg + (long long)e * DIM_D,
        Wu + wOff, bu + (long long)e * DIM_D, ws);
    down_kernel<<<grid2, 128, 0, stream>>>(
        ws, Wd + wOff, bd + (long long)e * DIM_H, rw, e, out);
  }
}